// Capsule_19387482375022
// MI455X (gfx1250) — compile-verified
//
#include <hip/hip_runtime.h>
#include <hip/hip_bf16.h>
#include <math.h>

// Problem constants (from reference): B=32, N=2048, D_in=8, C=32, D_out=16
#define NTOT   2048
#define CCAP   32
#define DIN    8
#define DOUT   16
#define BATCH  32
#define NPER   32              // n-values per K-chunk
#define NCHUNK (NTOT / NPER)   // 64 chunks
#define COLS   (CCAP * DOUT)   // 512 output columns
#define OUTEL  (BATCH * COLS)  // 16384 output elements

typedef __attribute__((ext_vector_type(2))) float v2f;
typedef __attribute__((ext_vector_type(8))) float v8f;

// ---------------------------------------------------------------------------
// Kernel 1: column-wise softmax of R over N.  R: [N, C] -> Rn: [N, C]
// One block (256 threads) per capsule column c.
// ---------------------------------------------------------------------------
__global__ __launch_bounds__(256)
void caps_softmax(const float* __restrict__ R, float* __restrict__ Rn) {
    const int c   = blockIdx.x;
    const int tid = threadIdx.x;
    __shared__ float red[256];

    float m = -INFINITY;
    for (int n = tid; n < NTOT; n += 256) m = fmaxf(m, R[n * CCAP + c]);
    red[tid] = m; __syncthreads();
    for (int s = 128; s > 0; s >>= 1) {
        if (tid < s) red[tid] = fmaxf(red[tid], red[tid + s]);
        __syncthreads();
    }
    m = red[0]; __syncthreads();

    float sum = 0.0f;
    for (int n = tid; n < NTOT; n += 256) sum += __expf(R[n * CCAP + c] - m);
    red[tid] = sum; __syncthreads();
    for (int s = 128; s > 0; s >>= 1) {
        if (tid < s) red[tid] += red[tid + s];
        __syncthreads();
    }
    const float inv = 1.0f / red[0];

    for (int n = tid; n < NTOT; n += 256)
        Rn[n * CCAP + c] = __expf(R[n * CCAP + c] - m) * inv;
}

// ---------------------------------------------------------------------------
// Kernel 2: WMMA GEMM over a K-chunk.
//   D[b, c*16+j] += sum_{n in chunk} sum_k x[b, n*8+k] * Rn[n,c]*W[n,c,j,k]
// Grid: (32 capsule tiles, 64 K-chunks); one wave32 per block.
// A (16x4 f32): lane<16 -> M=lane, K=+0,+1 ; lane>=16 -> M=lane-16, K=+2,+3
// B (4x16 f32): lane<16 -> N=lane, K=+0,+1 ; lane>=16 -> N=lane-16, K=+2,+3
// ---------------------------------------------------------------------------
__global__ __launch_bounds__(32)
void caps_gemm(const float* __restrict__ x, const float* __restrict__ W,
               const float* __restrict__ Rn, float* __restrict__ part) {
    const int t    = blockIdx.x;          // capsule tile == capsule c (16 cols)
    const int ch   = blockIdx.y;          // K-chunk
    const int lane = threadIdx.x;
    const int j    = lane & 15;           // column within tile / row M within tile
    const int hi   = lane >> 4;           // 0: K+0..1, 1: K+2..3
    const int kofs = hi * 2;

    v8f acc0 = {};                        // b = 0..15
    v8f acc1 = {};                        // b = 16..31

    const int n0 = ch * NPER;
    const float* wp  = W  + (size_t)n0 * (CCAP * DOUT * DIN) + t * (DOUT * DIN) + j * DIN + kofs;
    const float* xp0 = x  + (size_t)j        * (NTOT * DIN) + n0 * DIN + kofs;
    const float* xp1 = x  + (size_t)(j + 16) * (NTOT * DIN) + n0 * DIN + kofs;
    const float* rp  = Rn + n0 * CCAP + t;

    #pragma unroll 4
    for (int nn = 0; nn < NPER; ++nn) {
        const float rn = rp[nn * CCAP];   // wave-uniform routing weight

        v2f b0, b1, a00, a01, a10, a11;
        b0.x  = wp[0] * rn;  b0.y  = wp[1] * rn;     // K-step 0 (kbase=0)
        b1.x  = wp[4] * rn;  b1.y  = wp[5] * rn;     // K-step 1 (kbase=4)
        a00.x = xp0[0];      a00.y = xp0[1];
        a01.x = xp0[4];      a01.y = xp0[5];
        a10.x = xp1[0];      a10.y = xp1[1];
        a11.x = xp1[4];      a11.y = xp1[5];

        acc0 = __builtin_amdgcn_wmma_f32_16x16x4_f32(false, a00, false, b0,
                                                     (short)0, acc0, false, false);
        acc0 = __builtin_amdgcn_wmma_f32_16x16x4_f32(false, a01, false, b1,
                                                     (short)0, acc0, false, false);
        acc1 = __builtin_amdgcn_wmma_f32_16x16x4_f32(false, a10, false, b0,
                                                     (short)0, acc1, false, false);
        acc1 = __builtin_amdgcn_wmma_f32_16x16x4_f32(false, a11, false, b1,
                                                     (short)0, acc1, false, false);

        wp  += CCAP * DOUT * DIN;   // next n: +4096 floats
        xp0 += DIN;
        xp1 += DIN;
    }

    // C/D layout: VGPR v, lane -> row M = v + (lane<16 ? 0 : 8), col = lane&15
    float* p = part + (size_t)ch * OUTEL + t * DOUT + j;
    #pragma unroll
    for (int v = 0; v < 8; ++v) {
        const int b = v + hi * 8;
        p[(size_t)b        * COLS] = acc0[v];
        p[(size_t)(b + 16) * COLS] = acc1[v];
    }
}

// ---------------------------------------------------------------------------
// Kernel 3: reduce partials over chunks + squash.
// idx = b*512 + c*16 + j ; squash over the 16 j-lanes (contiguous lane groups).
// ---------------------------------------------------------------------------
__global__ __launch_bounds__(256)
void caps_reduce_squash(const float* __restrict__ part, float* __restrict__ out) {
    const int idx = blockIdx.x * 256 + threadIdx.x;   // 0..16383
    float s = 0.0f;
    #pragma unroll 8
    for (int ch = 0; ch < NCHUNK; ++ch)
        s += part[(size_t)ch * OUTEL + idx];

    float sq = s * s;
    // butterfly sum within each 16-lane (j) group of the wave32
    for (int m = 1; m < 16; m <<= 1)
        sq += __shfl_xor(sq, m, 32);
    sq += 1e-7f;
    const float scale = sqrtf(sq) / (1.0f + sq);
    out[idx] = scale * s;
}

// ---------------------------------------------------------------------------
extern "C" void kernel_launch(void* const* d_in, const int* in_sizes, int n_in,
                              void* d_out, int out_size, void* d_ws, size_t ws_size,
                              hipStream_t stream) {
    const float* x = (const float*)d_in[0];   // [32, 2048, 8]
    const float* W = (const float*)d_in[1];   // [1, 2048, 32, 16, 8]
    const float* R = (const float*)d_in[2];   // [1, 2048, 32]
    float* out = (float*)d_out;               // [32, 32, 16]

    float* Rn   = (float*)d_ws;               // 2048*32 floats   (256 KB)
    float* part = Rn + (size_t)NTOT * CCAP;   // 64*16384 floats  (4 MB)

    caps_softmax<<<CCAP, 256, 0, stream>>>(R, Rn);

    dim3 grid(CCAP, NCHUNK);                  // 32 x 64 = 2048 wave32 blocks
    caps_gemm<<<grid, 32, 0, stream>>>(x, W, Rn, part);

    caps_reduce_squash<<<OUTEL / 256, 256, 0, stream>>>(part, out);
}